// Contraction_21492016349382
// MI455X (gfx1250) — compile-verified
//
#include <hip/hip_runtime.h>
#include <hip/hip_bf16.h>

typedef __attribute__((ext_vector_type(2))) float v2f;
typedef __attribute__((ext_vector_type(8))) float v8f;

#define B_N   1024
#define C_N   128
#define ELL_N 16
#define EQ_N  3
#define E_N   10
#define P3_N  23
#define P2_N  5
#define M_N   768   // EQ*ELL*ELL
#define WX_N  48    // EQ*ELL
#define NT_N  32    // N-tile columns (2 WMMA tiles)

// ---------------------------------------------------------------------------
// Kernel 1: group batch indices by element (y is one-hot over E=10).
// Single block; serial stable partition by thread 0 (B=1024, negligible cost).
// Per-b output is independent of intra-group order -> deterministic output.
// ---------------------------------------------------------------------------
__global__ __launch_bounds__(1024) void group_kernel(const float* __restrict__ y,
                                                     int* __restrict__ blist,
                                                     int* __restrict__ counts) {
  __shared__ int es[B_N];
  int b = threadIdx.x;
  const float* yr = y + b * E_N;
  int best = 0; float bv = yr[0];
  for (int e = 1; e < E_N; ++e) { float v = yr[e]; if (v > bv) { bv = v; best = e; } }
  es[b] = best;
  __syncthreads();
  if (b == 0) {
    int cnt[E_N];
    for (int e = 0; e < E_N; ++e) cnt[e] = 0;
    for (int i = 0; i < B_N; ++i) {
      int e = es[i];
      blist[e * B_N + cnt[e]] = i;
      cnt[e]++;
    }
    for (int e = 0; e < E_N; ++e) counts[e] = cnt[e];
  }
}

// ---------------------------------------------------------------------------
// Kernel 2: one block per (e,c). Build A_ec (768x16, k folded) in LDS, then
// GEMM A_ec @ X(16 x n_e) via V_WMMA_F32_16X16X4_F32 (32-column N-tiles,
// X staged with global_load_async_to_lds_b64) with fused l2/l1 epilogue.
// ---------------------------------------------------------------------------
__global__ __launch_bounds__(256) void contract_kernel(
    const float* __restrict__ x,
    const float* __restrict__ U3,
    const float* __restrict__ U2,
    const float* __restrict__ U1,
    const float* __restrict__ wmax,
    const float* __restrict__ w2,
    const float* __restrict__ w1,
    const int* __restrict__ blist,
    const int* __restrict__ counts,
    float* __restrict__ out) {
  __shared__ float As[M_N][17];              // A_ec[m][i], stride-17 (bank-safe)
  __shared__ float A2s[M_N];                 // U2 . wn2 per m
  __shared__ float A1s[WX_N];                // U1 . wn1 per (w,x)
  __shared__ float wk3[P3_N];
  __shared__ float wk2[P2_N];
  __shared__ float wk1s;
  __shared__ __align__(16) float Xt[NT_N][18]; // Xt[n][i]; 72B rows (b64-aligned)
  __shared__ float t1buf[WX_N][NT_N];        // t1[w,x][n]

  const int tid = threadIdx.x;
  const int e = blockIdx.x / C_N;
  const int c = blockIdx.x % C_N;
  const int n_e = counts[e];

  // per-(e,c) learned weights
  if (tid < P3_N) wk3[tid] = wmax[(e * P3_N + tid) * C_N + c];
  if (tid >= 32 && tid < 32 + P2_N) wk2[tid - 32] = w2[(e * P2_N + (tid - 32)) * C_N + c];
  if (tid == 63) wk1s = w1[e * C_N + c];
  __syncthreads();

  // A_ec[m][i] = sum_k U3[m,i,k] * wk3[k]   (U3 flat: ((m*16)+i)*23 + k)
  for (int idx = tid; idx < M_N * ELL_N; idx += 256) {
    int m = idx >> 4, i = idx & 15;
    const float* u = U3 + (size_t)(m * ELL_N + i) * P3_N;
    float s = 0.f;
#pragma unroll
    for (int k = 0; k < P3_N; ++k) s += u[k] * wk3[k];
    As[m][i] = s;
  }
  for (int m = tid; m < M_N; m += 256) {
    const float* u = U2 + m * P2_N;
    float s = 0.f;
#pragma unroll
    for (int k = 0; k < P2_N; ++k) s += u[k] * wk2[k];
    A2s[m] = s;
  }
  if (tid < WX_N) A1s[tid] = U1[tid] * wk1s;
  __syncthreads();

  const int lane = tid & 31;
  const int half = lane >> 4;   // 0: lanes 0-15, 1: lanes 16-31
  const int ln   = lane & 15;
  const int wave = tid >> 5;    // 8 waves

  // async-stage geometry: thread t moves 2 floats (b64) of Xt[col][ip..ip+1]
  const int scol = tid >> 3;          // 0..31
  const int sip  = (tid & 7) * 2;     // 0,2,..,14

  for (int nt = 0; nt * NT_N < n_e; ++nt) {
    // stage X tile (32 columns x 16 i) directly into LDS via async DMA.
    // Padded columns clamp to batch row 0; their results are never stored.
    {
      int n = nt * NT_N + scol;
      int bsel = (n < n_e) ? blist[e * B_N + n] : 0;
      const float* gp = x + (size_t)bsel * (C_N * ELL_N) + c * ELL_N + sip;
      unsigned ldsaddr = (unsigned)(uintptr_t)(&Xt[scol][sip]);
      asm volatile("global_load_async_to_lds_b64 %0, %1, off"
                   :: "v"(ldsaddr), "v"(gp) : "memory");
    }
    asm volatile("s_wait_asynccnt 0" ::: "memory");
    __syncthreads();

    // 48 M-tiles (one (w,x) pair each; rows = v), 6 per wave, 2 N-subtiles
    for (int wx = wave; wx < WX_N; wx += 8) {
      v8f acc0 = {}, acc1 = {};
#pragma unroll
      for (int kk = 0; kk < 4; ++kk) {
        // A 16x4 layout: lanes 0-15 -> K=k0,k0+1 ; lanes 16-31 -> K=k0+2,k0+3
        int k0 = kk * 4 + half * 2;
        v2f a, b0, b1;
        a.x  = As[wx * 16 + ln][k0];
        a.y  = As[wx * 16 + ln][k0 + 1];
        b0.x = Xt[ln][k0];       b0.y = Xt[ln][k0 + 1];
        b1.x = Xt[16 + ln][k0];  b1.y = Xt[16 + ln][k0 + 1];
        acc0 = __builtin_amdgcn_wmma_f32_16x16x4_f32(false, a, false, b0,
                                                     (short)0, acc0, false, false);
        acc1 = __builtin_amdgcn_wmma_f32_16x16x4_f32(false, a, false, b1,
                                                     (short)0, acc1, false, false);
      }
      // epilogue: (out3 + A2) contracted over v using C/D lane layout
      // lane half==0 holds v=0..7 in acc[0..7]; half==1 holds v=8..15
      float p0 = 0.f, p1 = 0.f;
#pragma unroll
      for (int j = 0; j < 8; ++j) {
        int v = half * 8 + j;
        float a2 = A2s[wx * 16 + v];
        p0 += (acc0[j] + a2) * Xt[ln][v];
        p1 += (acc1[j] + a2) * Xt[16 + ln][v];
      }
      p0 += __shfl_xor(p0, 16, 32);   // combine the two halves
      p1 += __shfl_xor(p1, 16, 32);
      if (half == 0) {
        float a1 = A1s[wx];
        t1buf[wx][ln]      = p0 + a1;
        t1buf[wx][16 + ln] = p1 + a1;
      }
    }
    __syncthreads();

    // final l1 contraction over x: 96 threads -> (w, n)
    if (tid < EQ_N * NT_N) {
      int w = tid >> 5, n = tid & 31;
      float s = 0.f;
#pragma unroll
      for (int xx = 0; xx < 16; ++xx)
        s += t1buf[w * 16 + xx][n] * Xt[n][xx];
      int ng = nt * NT_N + n;
      if (ng < n_e) {
        int b = blist[e * B_N + ng];
        out[(size_t)b * (C_N * EQ_N) + c * EQ_N + w] = s;
      }
    }
    __syncthreads();
  }
}

extern "C" void kernel_launch(void* const* d_in, const int* in_sizes, int n_in,
                              void* d_out, int out_size, void* d_ws, size_t ws_size,
                              hipStream_t stream) {
  const float* x    = (const float*)d_in[0];
  const float* y    = (const float*)d_in[1];
  const float* U3   = (const float*)d_in[2];
  const float* U2   = (const float*)d_in[3];
  const float* U1   = (const float*)d_in[4];
  const float* wmax = (const float*)d_in[5];
  const float* w2   = (const float*)d_in[6];
  const float* w1   = (const float*)d_in[7];
  float* out = (float*)d_out;

  int* blist  = (int*)d_ws;            // E_N * B_N ints (40 KB)
  int* counts = blist + E_N * B_N;     // E_N ints

  hipLaunchKernelGGL(group_kernel, dim3(1), dim3(B_N), 0, stream,
                     y, blist, counts);
  hipLaunchKernelGGL(contract_kernel, dim3(E_N * C_N), dim3(256), 0, stream,
                     x, U3, U2, U1, wmax, w2, w1, blist, counts, out);
}